// Rule_4346506903625
// MI455X (gfx1250) — compile-verified
//
#include <hip/hip_runtime.h>
#include <stdint.h>

#define H 2048
#define W 2048
#define HW (H * W)
#define WMASK 2047
#define DT 1.0f
#define TEMP 0.1f
#define EPS 1e-8f

#define TILE 16
#define H2 20                    // tile + 2-halo staging region
#define NH2 (H2 * H2)            // 400
#define H1 18                    // tile + 1-halo (force'/momP/prop region)
#define NH1 (H1 * H1)            // 324
#define NBLK (HW / 256)          // 16384 tiles of 16x16

// ---------------------------------------------------------------------------
// CDNA5 async global->LDS copy (tracked by ASYNCcnt). Per-lane: copies 4 bytes
// from SADDR+VOFF into LDS at the per-lane LDS byte address.
// ---------------------------------------------------------------------------
__device__ __forceinline__ uint32_t lds_addr32(const void* p) {
    // Low 32 bits of a flat pointer into the LDS aperture == LDS byte offset.
    return (uint32_t)(uintptr_t)p;
}

__device__ __forceinline__ void async_g2l_b32(uint32_t lds_off, const float* base,
                                              uint32_t byte_off) {
    asm volatile("global_load_async_to_lds_b32 %0, %1, %2 offset:0"
                 :
                 : "v"(lds_off), "v"(byte_off), "s"(base)
                 : "memory");
}

__device__ __forceinline__ void async_wait0() {
    asm volatile("s_wait_asynccnt 0x0" ::: "memory");
}

// ---------------------------------------------------------------------------
// Fused kernel: s -> conv/force' -> momentum' -> softmax prop -> MOVES gather
// One 16x16 tile per block; all intermediates live in LDS.
// ---------------------------------------------------------------------------
__global__ void k_fused(const float* __restrict__ mass, const float* __restrict__ mom,
                        const float* __restrict__ force, const float* __restrict__ A,
                        const float* __restrict__ kern,
                        float* __restrict__ outMass, float* __restrict__ outMom,
                        float* __restrict__ forceOut,
                        float* __restrict__ partialA, float* __restrict__ partialB) {
    __shared__ float mT[NH2];    // original mass (never overwritten)
    __shared__ float aT[NH2];    // A, then s (in-place)
    __shared__ float fxT[NH2];   // force x, then force' x (in-place)
    __shared__ float fyT[NH2];   // force y, then force' y (in-place)
    __shared__ float mxT[NH2];   // momentum x, then momentum' x (in-place)
    __shared__ float myT[NH2];   // momentum y, then momentum' y (in-place)
    __shared__ float prT[NH2 * 9];
    __shared__ float kL[36];
    __shared__ float red[256];

    int tid = threadIdx.x;
    int bx = blockIdx.x & 127;
    int by = blockIdx.x >> 7;
    int x0 = bx * TILE, y0 = by * TILE;

    // ---- Phase 1: async-stage 6 input arrays over the 20x20 halo-2 region
    if (tid < 36) kL[tid] = kern[tid];
    for (int k = tid; k < NH2; k += 256) {
        int h = k / H2, w = k % H2;
        int gy = (y0 + h - 2) & WMASK;
        int gx = (x0 + w - 2) & WMASK;
        uint32_t voff = (uint32_t)((gy * W + gx) * 4);
        async_g2l_b32(lds_addr32(&mT[k]),  mass,       voff);
        async_g2l_b32(lds_addr32(&aT[k]),  A,          voff);
        async_g2l_b32(lds_addr32(&fxT[k]), force,      voff);
        async_g2l_b32(lds_addr32(&fyT[k]), force + HW, voff);
        async_g2l_b32(lds_addr32(&mxT[k]), mom,        voff);
        async_g2l_b32(lds_addr32(&myT[k]), mom + HW,   voff);
    }
    async_wait0();
    __syncthreads();

    // ---- Phase 2: s = A*mass + |force|, in-place over aT (all 400 cells)
    for (int k = tid; k < NH2; k += 256) {
        float ffx = fxT[k], ffy = fyT[k];
        aT[k] = aT[k] * mT[k] + sqrtf(ffx * ffx + ffy * ffy);
    }
    __syncthreads();

    // ---- Phase 3: halo-1 ring: conv -> force' -> momentum' -> softmax props
    float asum = 0.f;
#pragma unroll
    for (int j = 0; j < 36; ++j) asum += fabsf(kL[j]);
    float kc = 1.0f / asum;

    const float SQ2 = 0.70710678118654752f;
    const float dirx[9] = { SQ2, 0.f, -SQ2, 1.f, 0.f, -1.f,  SQ2,  0.f, -SQ2 };
    const float diry[9] = { SQ2, 1.f,  SQ2, 0.f, 0.f,  0.f, -SQ2, -1.f, -SQ2 };

    for (int k = tid; k < NH1; k += 256) {
        int h = k / H1 + 1, w = k % H1 + 1;   // halo-2 coords in [1,18]
        int c = h * H2 + w;

        float nf0 = 0.f, nf1 = 0.f, nf2 = 0.f, nf3 = 0.f;
#pragma unroll
        for (int r = 0; r < 3; ++r)
#pragma unroll
            for (int cc = 0; cc < 3; ++cc) {
                float sv = aT[(h - 1 + r) * H2 + (w - 1 + cc)];
                int wi = r * 3 + cc;
                nf0 += sv * kL[0 * 9 + wi];
                nf1 += sv * kL[1 * 9 + wi];
                nf2 += sv * kL[2 * 9 + wi];
                nf3 += sv * kL[3 * 9 + wi];
            }
        float n0 = nf0 + nf1;
        float n1 = 0.5f * (nf2 + nf3);

        float fx = fxT[c], fy = fyT[c];
        float nfx = fx + (n0 - fx) * kc;
        float nfy = fy + (n1 - fy) * kc;
        fxT[c] = nfx;
        fyT[c] = nfy;

        float m = mT[c];
        bool dead = m < EPS;
        float mpx = dead ? 0.f : mxT[c] + nfx * DT;
        float mpy = dead ? 0.f : myT[c] + nfy * DT;
        mxT[c] = mpx;
        myT[c] = mpy;

        float inv = dead ? 0.f : (1.0f / m);
        float vx = mpx * inv;
        float vy = mpy * inv;

        float t[9];
        float tmax = -3.402823466e38f;
#pragma unroll
        for (int d = 0; d < 9; ++d) {
            t[d] = TEMP * (dirx[d] * vx + diry[d] * vy);
            tmax = fmaxf(tmax, t[d]);
        }
        float den = 0.f, e[9];
#pragma unroll
        for (int d = 0; d < 9; ++d) { e[d] = expf(t[d] - tmax); den += e[d]; }
        float rden = 1.0f / den;
#pragma unroll
        for (int d = 0; d < 9; ++d) prT[c * 9 + d] = e[d] * rden;
    }
    __syncthreads();

    // ---- Phase 4: per interior pixel: outputs + MOVES gather
    // Neighbor at (dy,dx) contributes via exactly one direction index i.
    const int dyv[9] = { 1, 1, 1, 0, 0, 0, -1, -1, -1 };
    const int dxv[9] = { 1, 0, -1, 1, 0, -1, 1, 0, -1 };
    int ty = tid >> 4, tx = tid & 15;
    int c2 = (ty + 2) * H2 + (tx + 2);
    int gi = (y0 + ty) * W + (x0 + tx);

    forceOut[gi]      = fxT[c2];
    forceOut[gi + HW] = fyT[c2];

    float nm = 0.f, nmx = 0.f, nmy = 0.f;
#pragma unroll
    for (int i = 0; i < 9; ++i) {
        int n = (ty + 2 + dyv[i]) * H2 + (tx + 2 + dxv[i]);
        float p = prT[n * 9 + i];
        nm  += mT[n]  * p;
        nmx += mxT[n] * p;
        nmy += myT[n] * p;
    }
    outMass[gi]     = nm;
    outMom[gi]      = nmx;
    outMom[gi + HW] = nmy;

    // ---- deterministic block reductions: sum(mass), sum(new_mass)
    red[tid] = mT[c2];
    __syncthreads();
#pragma unroll
    for (int off = 128; off > 0; off >>= 1) {
        if (tid < off) red[tid] += red[tid + off];
        __syncthreads();
    }
    if (tid == 0) partialA[blockIdx.x] = red[0];
    __syncthreads();

    red[tid] = nm;
    __syncthreads();
#pragma unroll
    for (int off = 128; off > 0; off >>= 1) {
        if (tid < off) red[tid] += red[tid + off];
        __syncthreads();
    }
    if (tid == 0) partialB[blockIdx.x] = red[0];
}

// ---------------------------------------------------------------------------
// K4: reduce both partial arrays; scale = mass_tot / sum(new_mass)
// ---------------------------------------------------------------------------
__global__ void k4_reduce(const float* __restrict__ pA, const float* __restrict__ pB,
                          float* __restrict__ scale) {
    __shared__ float red[256];
    int tid = threadIdx.x;

    float a = 0.f;
    for (int i = tid; i < NBLK; i += 256) a += pA[i];
    red[tid] = a;
    __syncthreads();
#pragma unroll
    for (int off = 128; off > 0; off >>= 1) {
        if (tid < off) red[tid] += red[tid + off];
        __syncthreads();
    }
    float massTot = red[0];
    __syncthreads();

    float b = 0.f;
    for (int i = tid; i < NBLK; i += 256) b += pB[i];
    red[tid] = b;
    __syncthreads();
#pragma unroll
    for (int off = 128; off > 0; off >>= 1) {
        if (tid < off) red[tid] += red[tid + off];
        __syncthreads();
    }
    if (tid == 0) scale[0] = massTot / red[0];
}

// ---------------------------------------------------------------------------
// K5: new_mass *= scale
// ---------------------------------------------------------------------------
__global__ void k5_scale(float* __restrict__ outMass, const float* __restrict__ scale) {
    int i = blockIdx.x * 256 + threadIdx.x;
    outMass[i] *= scale[0];
}

// ---------------------------------------------------------------------------
extern "C" void kernel_launch(void* const* d_in, const int* in_sizes, int n_in,
                              void* d_out, int out_size, void* d_ws, size_t ws_size,
                              hipStream_t stream) {
    (void)in_sizes; (void)n_in; (void)out_size; (void)ws_size;

    const float* mass  = (const float*)d_in[0];
    const float* mom   = (const float*)d_in[1];
    const float* force = (const float*)d_in[2];
    const float* A     = (const float*)d_in[3];
    const float* kern  = (const float*)d_in[4];

    float* out      = (float*)d_out;
    float* outMass  = out;              // [0, HW)
    float* outMom   = out + HW;         // [HW, 3HW)
    float* outForce = out + 3 * HW;     // [3HW, 5HW)

    float* ws    = (float*)d_ws;
    float* pA    = ws;                  // NBLK
    float* pB    = pA + NBLK;           // NBLK
    float* scale = pB + NBLK;           // 1

    dim3 blk(256);
    k_fused<<<NBLK, blk, 0, stream>>>(mass, mom, force, A, kern,
                                      outMass, outMom, outForce, pA, pB);
    k4_reduce<<<1, blk, 0, stream>>>(pA, pB, scale);
    k5_scale<<<NBLK, blk, 0, stream>>>(outMass, scale);
}